// Masker_22686017258186
// MI455X (gfx1250) — compile-verified
//
#include <hip/hip_runtime.h>
#include <stdint.h>

// ---------------- problem constants (from reference) ----------------
#define B       64
#define C       3
#define HW      262144            // 512*512
#define CHW     786432            // C*HW
#define K_UPPER 196608

// ---------------- workspace layout (bytes) ----------------
#define OFF_MN   0                         // 192 u32 (order-encoded float min)
#define OFF_MX   1024                      // 192 u32 (order-encoded float max)
#define OFF_SEL  2048                      // B * 8 u32: bin1,below1,bin2,totalBelow,remaining,tieCtr
#define OFF_H1   8192                      // B*2048 u32 coarse histogram
#define OFF_H2   (OFF_H1 + B*2048*4)       // B*2048 u32 fine histogram
#define WS_USED  (OFF_H2 + B*2048*4)       // ~1.01 MB

#define MM_SLICES 8
#define BPB       64                       // histogram blocks per batch
#define CHUNK     1024                     // floats staged per block per iteration (4KB)

typedef __attribute__((ext_vector_type(2))) float v2f;
typedef __attribute__((ext_vector_type(8))) float v8f;

// ---------------- order-preserving float <-> uint ----------------
__device__ __forceinline__ uint32_t f2ord(float f) {
    uint32_t u = __float_as_uint(f);
    return (u & 0x80000000u) ? ~u : (u | 0x80000000u);
}
__device__ __forceinline__ float ord2f(uint32_t k) {
    uint32_t u = (k & 0x80000000u) ? (k & 0x7FFFFFFFu) : ~k;
    return __uint_as_float(u);
}

// ---------------- CDNA5 async global->LDS path ----------------
__device__ __forceinline__ void async_load_b128(uint32_t lds_addr, const void* gaddr) {
    // VDST = per-lane LDS byte address, VADDR = 64-bit global address, no SADDR.
    asm volatile("global_load_async_to_lds_b128 %0, %1, off"
                 :: "v"(lds_addr), "v"((unsigned long long)(uintptr_t)gaddr)
                 : "memory");
}
__device__ __forceinline__ void wait_async_le1() {
#if __has_builtin(__builtin_amdgcn_s_wait_asynccnt)
    __builtin_amdgcn_s_wait_asynccnt(1);
#else
    asm volatile("s_wait_asynccnt 0x1" ::: "memory");
#endif
}
__device__ __forceinline__ void wait_async_le0() {
#if __has_builtin(__builtin_amdgcn_s_wait_asynccnt)
    __builtin_amdgcn_s_wait_asynccnt(0);
#else
    asm volatile("s_wait_asynccnt 0x0" ::: "memory");
#endif
}

// ---------------- tiny init: encoded min/max identities ----------------
__global__ void __launch_bounds__(256) k_init(uint32_t* ws) {
    int t = threadIdx.x;
    if (t < B * C) {
        ws[OFF_MN/4 + t] = 0xFFFFFFFFu;   // +inf in encoded order
        ws[OFF_MX/4 + t] = 0u;            // -inf in encoded order
    }
}

// ---------------- per-(b,c) min/max of x[...,:HW-1] ----------------
__global__ void __launch_bounds__(256) k_minmax(const float* __restrict__ x,
                                                uint32_t* __restrict__ ws) {
    int blk   = blockIdx.x;               // B*C*MM_SLICES blocks
    int slice = blk % MM_SLICES;
    int bc    = blk / MM_SLICES;
    const int sliceLen = HW / MM_SLICES;  // 32768
    size_t base = (size_t)bc * HW + (size_t)slice * sliceLen;
    int t = threadIdx.x;
    int lastLocal = (HW - 1) - slice * sliceLen;   // exclude channel's last element
    float mn = 3.4e38f, mx = -3.4e38f;
    for (int i = t * 4; i < sliceLen; i += 256 * 4) {
        float4 v = *(const float4*)(x + base + i);
        float a[4] = {v.x, v.y, v.z, v.w};
        #pragma unroll
        for (int k = 0; k < 4; ++k) {
            if (i + k != lastLocal) { mn = fminf(mn, a[k]); mx = fmaxf(mx, a[k]); }
        }
    }
    __shared__ float smn[256], smx[256];
    smn[t] = mn; smx[t] = mx; __syncthreads();
    for (int off = 128; off > 0; off >>= 1) {
        if (t < off) { smn[t] = fminf(smn[t], smn[t+off]); smx[t] = fmaxf(smx[t], smx[t+off]); }
        __syncthreads();
    }
    if (t == 0) {
        atomicMin((unsigned int*)&ws[OFF_MN/4 + bc], f2ord(smn[0]));
        atomicMax((unsigned int*)&ws[OFF_MX/4 + bc], f2ord(smx[0]));
    }
}

// ---------------- histogram passes (async-LDS double-buffered stream) ----------------
__global__ void __launch_bounds__(256) k_hist(const float* __restrict__ grad,
                                              uint32_t* __restrict__ ws, int level) {
    int blk = blockIdx.x;                 // B*BPB blocks
    int b   = blk / BPB;
    int sub = blk % BPB;
    const int perBlock = CHW / BPB;       // 12288
    size_t base = (size_t)b * CHW + (size_t)sub * perBlock;
    int t = threadIdx.x;

    __shared__ uint32_t hist[2048];
    __shared__ __align__(16) float buf[2][CHUNK];
    for (int i = t; i < 2048; i += 256) hist[i] = 0;

    uint32_t bin1 = 0;
    uint32_t* gh;
    if (level == 0) gh = ws + OFF_H1/4 + b * 2048;
    else { bin1 = ws[OFF_SEL/4 + b*8 + 0]; gh = ws + OFF_H2/4 + b * 2048; }
    __syncthreads();

    uint32_t lds0 = (uint32_t)(uintptr_t)&buf[0][t * 4];
    uint32_t lds1 = (uint32_t)(uintptr_t)&buf[1][t * 4];
    const float* g0 = grad + base + t * 4;
    const int nIter = perBlock / CHUNK;   // 12

    async_load_b128(lds0, g0);            // prologue: chunk 0
    for (int it = 0; it < nIter; ++it) {
        if (it + 1 < nIter) {
            async_load_b128(((it + 1) & 1) ? lds1 : lds0, g0 + (size_t)(it + 1) * CHUNK);
            wait_async_le1();             // oldest (chunk it) complete; in-order done
        } else {
            wait_async_le0();
        }
        float4 v = *(float4*)&buf[it & 1][t * 4];   // ds_load_b128 of own staged slice
        float a[4] = {v.x, v.y, v.z, v.w};
        #pragma unroll
        for (int k = 0; k < 4; ++k) {
            uint32_t ab = __float_as_uint(a[k]) & 0x7FFFFFFFu;
            if (level == 0) {
                atomicAdd(&hist[ab >> 20], 1u);                       // bits[30:20]
            } else if ((ab >> 20) == bin1) {
                atomicAdd(&hist[(ab >> 9) & 0x7FFu], 1u);             // bits[19:9]
            }
        }
    }
    __syncthreads();
    for (int i = t; i < 2048; i += 256) {
        uint32_t h = hist[i];
        if (h) atomicAdd((unsigned int*)&gh[i], h);
    }
}

// ---------------- WMMA prefix-scan: locate the K-th boundary bin ----------------
// One wave32 per batch. Bin order: bin = col*16 + row (col 0..127, row 0..15).
// For each 16-column chunk, P = L x M (L lower-triangular ones, 16x16) computed
// with 4 chained V_WMMA_F32_16X16X4_F32 gives exact per-column inclusive scans
// (all counts < 2^24, exactly representable in f32).
__global__ void __launch_bounds__(32) k_scan(uint32_t* __restrict__ ws, int level) {
    int b = blockIdx.x;
    int l = threadIdx.x;
    const uint32_t* h = ws + ((level == 0) ? OFF_H1/4 : OFF_H2/4) + b * 2048;
    uint32_t* sel = ws + OFF_SEL/4 + b * 8;
    uint32_t Kt = (level == 0) ? (uint32_t)K_UPPER : ((uint32_t)K_UPPER - sel[1]);

    __shared__ float colScan[2048];
    int n    = l & 15;      // column index within chunk (A: m index)
    int half = l >> 4;

    for (int cc = 0; cc < 8; ++cc) {      // uniform loop: EXEC all-ones at WMMA
        v8f acc = {};
        #pragma unroll
        for (int s = 0; s < 4; ++s) {
            v2f A, Bv;
            #pragma unroll
            for (int v = 0; v < 2; ++v) {
                int k = 4 * s + v + 2 * half;          // K index per A/B layout
                A[v]  = (k <= n) ? 1.0f : 0.0f;        // L[m=n][k]
                Bv[v] = (float)h[256 * cc + 16 * n + k];   // M[k][col=n]
            }
            acc = __builtin_amdgcn_wmma_f32_16x16x4_f32(
                      false, A, false, Bv, (short)0, acc, false, false);
        }
        // D layout: lane l, vgpr g -> P[m = g + 8*half][n = l&15]
        #pragma unroll
        for (int g = 0; g < 8; ++g) {
            colScan[(16 * cc + n) * 16 + g + 8 * half] = acc[g];
        }
    }
    __syncthreads();

    if (l == 0) {
        uint32_t run = 0, colOff = 0;
        int cstar = -1;
        for (int col = 0; col < 128; ++col) {
            uint32_t tot = (uint32_t)(colScan[col * 16 + 15] + 0.5f);
            if (cstar < 0 && run + tot >= Kt) { cstar = col; colOff = run; }
            run += tot;
        }
        uint32_t need = Kt - colOff;
        uint32_t bin = (uint32_t)cstar * 16, below = colOff;
        for (int r = 0; r < 16; ++r) {
            uint32_t inc = (uint32_t)(colScan[cstar * 16 + r] + 0.5f);
            if (inc >= need) {
                bin   = (uint32_t)cstar * 16 + r;
                below = colOff + (r > 0 ? (uint32_t)(colScan[cstar * 16 + r - 1] + 0.5f) : 0u);
                break;
            }
        }
        if (level == 0) { sel[0] = bin; sel[1] = below; }
        else {
            uint32_t tb = sel[1] + below;
            sel[2] = bin;
            sel[3] = tb;
            sel[4] = (uint32_t)K_UPPER - tb;   // tie quota
            sel[5] = 0u;                       // tie counter
        }
    }
}

// ---------------- fused fill pass ----------------
__global__ void __launch_bounds__(256) k_fill(const float* __restrict__ x,
                                              const float* __restrict__ grad,
                                              float* __restrict__ out,
                                              uint32_t* __restrict__ ws) {
    size_t gi = ((size_t)blockIdx.x * 256 + threadIdx.x) * 4;
    int b   = (int)(gi / CHW);            // uniform per block (block spans 1024 elems)
    int rem = (int)(gi % CHW);
    int c   = rem / HW;                   // uniform per block (HW % 1024 == 0)

    uint32_t* sel = ws + OFF_SEL/4 + b * 8;
    uint32_t thresh22  = (sel[0] << 11) | sel[2];
    uint32_t remaining = sel[4];
    unsigned int* tie  = (unsigned int*)&sel[5];
    float lo = ord2f(ws[OFF_MN/4 + b * C + c]);
    float hi = ord2f(ws[OFF_MX/4 + b * C + c]);

    float4 xv = *(const float4*)(x + gi);
    float4 gv = *(const float4*)(grad + gi);
    float xo[4] = {xv.x, xv.y, xv.z, xv.w};
    float ga[4] = {gv.x, gv.y, gv.z, gv.w};
    #pragma unroll
    for (int k = 0; k < 4; ++k) {
        uint32_t ab  = __float_as_uint(ga[k]) & 0x7FFFFFFFu;
        uint32_t key = ab >> 9;
        bool m = key < thresh22;
        if (!m && key == thresh22) m = (atomicAdd(tie, 1u) < remaining);
        if (m) {
            uint32_t hsh = (uint32_t)(gi + (size_t)k) * 0x9E3779B9u ^ (uint32_t)b * 0x85EBCA6Bu;
            hsh ^= hsh >> 16; hsh *= 0x7FEB352Du; hsh ^= hsh >> 15; hsh *= 0x846CA68Bu; hsh ^= hsh >> 16;
            float u = (float)(hsh >> 8) * (1.0f / 16777216.0f);   // [0,1)
            xo[k] = lo + u * (hi - lo);
        }
    }
    float4 o; o.x = xo[0]; o.y = xo[1]; o.z = xo[2]; o.w = xo[3];
    *(float4*)(out + gi) = o;
}

extern "C" void kernel_launch(void* const* d_in, const int* in_sizes, int n_in,
                              void* d_out, int out_size, void* d_ws, size_t ws_size,
                              hipStream_t stream) {
    (void)in_sizes; (void)n_in; (void)out_size; (void)ws_size;
    const float* x    = (const float*)d_in[0];
    const float* grad = (const float*)d_in[1];
    float* out        = (float*)d_out;
    uint32_t* ws      = (uint32_t*)d_ws;

    hipMemsetAsync(d_ws, 0, WS_USED, stream);
    k_init  <<<1, 256, 0, stream>>>(ws);
    k_minmax<<<B * C * MM_SLICES, 256, 0, stream>>>(x, ws);
    k_hist  <<<B * BPB, 256, 0, stream>>>(grad, ws, 0);
    k_scan  <<<B, 32, 0, stream>>>(ws, 0);
    k_hist  <<<B * BPB, 256, 0, stream>>>(grad, ws, 1);
    k_scan  <<<B, 32, 0, stream>>>(ws, 1);
    k_fill  <<<(unsigned)((size_t)B * CHW / 4 / 256), 256, 0, stream>>>(x, grad, out, ws);
}